// GIN_GINE_13898514170647
// MI455X (gfx1250) — compile-verified
//
#include <hip/hip_runtime.h>
#include <hip/hip_bf16.h>

#define N_NODES 50000
#define N_EDGES 800000
#define N_GRAPHS 512
#define HID 64
#define F_NODE 9
#define F_EDGE 3
#define N_TASKS 12
#define BN_EPS 1e-5f

typedef __attribute__((ext_vector_type(2))) float v2f;
typedef __attribute__((ext_vector_type(8))) float v8f;

// ---------------------------------------------------------------------------
// zero fill
__global__ void k_zero(float* __restrict__ p, int n) {
    int i = blockIdx.x * blockDim.x + threadIdx.x;
    if (i < n) p[i] = 0.0f;
}

// ---------------------------------------------------------------------------
// layer 0 message+scatter: agg9[dst] += relu(x[src] + edge_attr @ we0 + be0)
// one thread per edge, 9 channels
__global__ void k_scatter0(const float* __restrict__ x,
                           const float* __restrict__ ea,
                           const float* __restrict__ we0,   // [3,9]
                           const float* __restrict__ be0,   // [9]
                           const int* __restrict__ src,
                           const int* __restrict__ dst,
                           float* __restrict__ agg9) {
    int e = blockIdx.x * blockDim.x + threadIdx.x;
    if (e >= N_EDGES) return;
    int s = src[e], d = dst[e];
    float e0 = ea[e * 3 + 0], e1 = ea[e * 3 + 1], e2 = ea[e * 3 + 2];
    const float* xp = x + (long)s * F_NODE;
    float* ap = agg9 + (long)d * F_NODE;
#pragma unroll
    for (int c = 0; c < F_NODE; ++c) {
        float et = e0 * we0[0 * F_NODE + c] + e1 * we0[1 * F_NODE + c] +
                   e2 * we0[2 * F_NODE + c] + be0[c];
        float m = fmaxf(xp[c] + et, 0.0f);
        unsafeAtomicAdd(ap + c, m);
    }
}

// ---------------------------------------------------------------------------
// hidden-layer message+scatter: 8 threads per edge, 8 contiguous channels each
// gine: agg[dst] += relu(h[src] + edge_attr @ we + be)
// gin : agg[dst] += h[src]
template <bool GINE>
__global__ void k_scatter64(const float* __restrict__ h,
                            const float* __restrict__ ea,
                            const float* __restrict__ we,   // [3,64]
                            const float* __restrict__ be,   // [64]
                            const int* __restrict__ src,
                            const int* __restrict__ dst,
                            float* __restrict__ agg) {
    long gid = (long)blockIdx.x * blockDim.x + threadIdx.x;
    int e = (int)(gid >> 3);
    if (e >= N_EDGES) return;
    int cb = ((int)gid & 7) * 8;
    int s = src[e], d = dst[e];
    const float* hp = h + (long)s * HID + cb;
    float* ap = agg + (long)d * HID + cb;
    if (GINE) {
        float e0 = ea[e * 3 + 0], e1 = ea[e * 3 + 1], e2 = ea[e * 3 + 2];
#pragma unroll
        for (int c = 0; c < 8; ++c) {
            float et = e0 * we[0 * HID + cb + c] + e1 * we[1 * HID + cb + c] +
                       e2 * we[2 * HID + cb + c] + be[cb + c];
            float m = fmaxf(hp[c] + et, 0.0f);
            unsafeAtomicAdd(ap + c, m);
        }
    } else {
#pragma unroll
        for (int c = 0; c < 8; ++c) unsafeAtomicAdd(ap + c, hp[c]);
    }
}

// ---------------------------------------------------------------------------
// layer-0 first linear (K=9, VALU; trivial FLOPs): t = relu((x+agg9)@w1_0+b1_0)
__global__ void k_dense0(const float* __restrict__ x,
                         const float* __restrict__ agg9,
                         const float* __restrict__ w,   // [9,64]
                         const float* __restrict__ b,   // [64]
                         float* __restrict__ out) {
    long gid = (long)blockIdx.x * blockDim.x + threadIdx.x;
    int r = (int)(gid >> 6);
    int c = (int)gid & 63;
    if (r >= N_NODES) return;
    float acc = b[c];
#pragma unroll
    for (int k = 0; k < F_NODE; ++k)
        acc += (x[(long)r * F_NODE + k] + agg9[(long)r * F_NODE + k]) * w[k * HID + c];
    out[(long)r * HID + c] = fmaxf(acc, 0.0f);
}

// ---------------------------------------------------------------------------
// WMMA f32 GEMM: out[N,64] = act( (A1 (+A2)) @ W[64,64] + bias )
// one wave -> 16 rows x 64 cols tile, K stepped by 4 via V_WMMA_F32_16X16X4_F32
// HAS_A2 is a template param so the hot K-loop is branch-free.
template <bool HAS_A2, bool RELU>
__global__ void k_mm64_wmma(const float* __restrict__ A1,
                            const float* __restrict__ A2,
                            const float* __restrict__ W,    // [64,64] row-major
                            const float* __restrict__ bias, // [64]
                            float* __restrict__ out,
                            int nrows) {
    int lane = threadIdx.x & 31;
    int wave = threadIdx.x >> 5;                 // 8 waves / block
    int row0 = (blockIdx.x * 8 + wave) * 16;
    if (row0 >= nrows) return;                   // wave-uniform exit: EXEC stays all-1s
    int m  = lane & 15;
    int kq = lane >> 4;                          // 0 | 1 (half-wave)
    int row = row0 + m;
    int rc  = row < nrows ? row : nrows - 1;     // clamp OOB A-loads
    const v2f* a1 = (const v2f*)(A1 + (long)rc * HID);
    const v2f* a2 = HAS_A2 ? (const v2f*)(A2 + (long)rc * HID) : nullptr;

    v8f acc[4] = {v8f{}, v8f{}, v8f{}, v8f{}};
#pragma unroll
    for (int k0 = 0; k0 < HID; k0 += 4) {
        int k = k0 + 2 * kq;                     // K pair index for this half-wave
        v2f a = a1[k >> 1];
        if (HAS_A2) a += a2[k >> 1];
#pragma unroll
        for (int t = 0; t < 4; ++t) {
            int n = t * 16 + m;
            v2f b;
            b.x = W[(k + 0) * HID + n];
            b.y = W[(k + 1) * HID + n];
            acc[t] = __builtin_amdgcn_wmma_f32_16x16x4_f32(
                false, a, false, b, (short)0, acc[t], false, false);
        }
    }
    // D layout: acc[t][r] = D[r + 8*kq][t*16 + m]
    bool full = (row0 + 16) <= nrows;            // wave-uniform fast path
#pragma unroll
    for (int t = 0; t < 4; ++t) {
        int n = t * 16 + m;
        float bv = bias[n];
#pragma unroll
        for (int r = 0; r < 8; ++r) {
            int rr = row0 + r + 8 * kq;
            if (full || rr < nrows) {
                float v = acc[t][r] + bv;
                if (RELU) v = fmaxf(v, 0.0f);
                out[(long)rr * HID + n] = v;
            }
        }
    }
}

// ---------------------------------------------------------------------------
// BN stats: stats[0:64]=sum, stats[64:128]=sum of squares (over nodes)
__global__ void k_bn_stats(const float* __restrict__ h, float* __restrict__ stats) {
    __shared__ float s1[256];
    __shared__ float s2[256];
    int tid = threadIdx.x;
    int c = tid & 63, rl = tid >> 6;             // 4 row-groups
    int base = blockIdx.x * 1024;
    int lim = base + 1024 < N_NODES ? base + 1024 : N_NODES;
    float a = 0.0f, b = 0.0f;
    for (int r = base + rl; r < lim; r += 4) {
        float v = h[(long)r * HID + c];
        a += v;
        b += v * v;
    }
    s1[tid] = a; s2[tid] = b;
    __syncthreads();
    if (rl == 0) {
        float sa = s1[c] + s1[64 + c] + s1[128 + c] + s1[192 + c];
        float sb = s2[c] + s2[64 + c] + s2[128 + c] + s2[192 + c];
        unsafeAtomicAdd(stats + c, sa);
        unsafeAtomicAdd(stats + 64 + c, sb);
    }
}

// BN apply (training mode, biased variance) + ReLU, in place
__global__ void k_bn_apply(float* __restrict__ h,
                           const float* __restrict__ stats,
                           const float* __restrict__ gamma,
                           const float* __restrict__ beta) {
    long gid = (long)blockIdx.x * blockDim.x + threadIdx.x;
    int r = (int)(gid >> 6);
    int c = (int)gid & 63;
    if (r >= N_NODES) return;
    float invn = 1.0f / (float)N_NODES;
    float mu = stats[c] * invn;
    float var = stats[64 + c] * invn - mu * mu;
    float sc = rsqrtf(var + BN_EPS) * gamma[c];
    float v = (h[(long)r * HID + c] - mu) * sc + beta[c];
    h[(long)r * HID + c] = fmaxf(v, 0.0f);
}

// ---------------------------------------------------------------------------
// graph pooling: pooled[g][0:64]=segment_sum, pooled[g][64:128]=segment_max
// h >= 0 after BN+ReLU, so int-bitpattern atomicMax with 0-init is exact and
// also implements the reference's empty-segment -> 0 guard.
__global__ void k_pool(const float* __restrict__ h,
                       const int* __restrict__ batch,
                       float* __restrict__ pooled) {
    long gid = (long)blockIdx.x * blockDim.x + threadIdx.x;
    int r = (int)(gid >> 6);
    int c = (int)gid & 63;
    if (r >= N_NODES) return;
    int g = batch[r];
    float v = h[(long)r * HID + c];
    unsafeAtomicAdd(pooled + (long)g * 128 + c, v);
    atomicMax((int*)(pooled + (long)g * 128 + 64 + c), __float_as_int(v));
}

// classifier: out[g,t] = pooled[g,:] @ wc + bc
__global__ void k_classify(const float* __restrict__ pooled,
                           const float* __restrict__ wc,   // [128,12]
                           const float* __restrict__ bc,   // [12]
                           float* __restrict__ out) {
    int gid = blockIdx.x * blockDim.x + threadIdx.x;
    if (gid >= N_GRAPHS * N_TASKS) return;
    int g = gid / N_TASKS, t = gid % N_TASKS;
    float acc = bc[t];
#pragma unroll 8
    for (int k = 0; k < 2 * HID; ++k) acc += pooled[(long)g * 128 + k] * wc[k * N_TASKS + t];
    out[gid] = acc;
}

// ---------------------------------------------------------------------------
extern "C" void kernel_launch(void* const* d_in, const int* in_sizes, int n_in,
                              void* d_out, int out_size, void* d_ws, size_t ws_size,
                              hipStream_t stream) {
    const float* x     = (const float*)d_in[0];
    const int*   eidx  = (const int*)d_in[1];
    const float* ea    = (const float*)d_in[2];
    const int*   batch = (const int*)d_in[3];
    const float* w1_0  = (const float*)d_in[4];
    const float* b1_0  = (const float*)d_in[5];
    const float* w2_0  = (const float*)d_in[6];
    const float* b2_0  = (const float*)d_in[7];
    const float* we0   = (const float*)d_in[8];
    const float* be0   = (const float*)d_in[9];
    const float* w1    = (const float*)d_in[10];
    const float* b1    = (const float*)d_in[11];
    const float* w2    = (const float*)d_in[12];
    const float* b2    = (const float*)d_in[13];
    const float* we    = (const float*)d_in[14];
    const float* be    = (const float*)d_in[15];
    const float* gam   = (const float*)d_in[16];
    const float* bet   = (const float*)d_in[17];
    const float* wc    = (const float*)d_in[18];
    const float* bc    = (const float*)d_in[19];
    float* out = (float*)d_out;

    const int* src = eidx;
    const int* dst = eidx + N_EDGES;

    // workspace carve-up
    float* ws = (float*)d_ws;
    const long NH = (long)N_NODES * HID;
    float* hA     = ws;                 // [N,64]
    float* hB     = hA + NH;            // [N,64]
    float* tmp    = hB + NH;            // [N,64]
    float* agg    = tmp + NH;           // [N,64] (first N*9 used as agg9)
    float* stats  = agg + NH;           // [128]
    float* pooled = stats + 128;        // [512,128]

    const int TB = 256;
    auto blks = [](long n, int tb) { return (int)((n + tb - 1) / tb); };
    const int mm_blocks = blks(N_NODES, 128);   // 8 waves x 16 rows per block

    // ---- layer 0 (GINE, F_NODE channels) ----
    k_zero<<<blks((long)N_NODES * F_NODE, TB), TB, 0, stream>>>(agg, N_NODES * F_NODE);
    k_scatter0<<<blks(N_EDGES, TB), TB, 0, stream>>>(x, ea, we0, be0, src, dst, agg);
    k_dense0<<<blks(NH, TB), TB, 0, stream>>>(x, agg, w1_0, b1_0, tmp);
    k_mm64_wmma<false, false><<<mm_blocks, TB, 0, stream>>>(tmp, nullptr, w2_0, b2_0, hA, N_NODES);
    k_zero<<<1, 128, 0, stream>>>(stats, 128);
    k_bn_stats<<<blks(N_NODES, 1024), TB, 0, stream>>>(hA, stats);
    k_bn_apply<<<blks(NH, TB), TB, 0, stream>>>(hA, stats, gam, bet);

    // ---- layers 1..4 ----
    float* hcur = hA;
    float* hnext = hB;
    for (int i = 0; i < 4; ++i) {
        int eix = i / 2;
        k_zero<<<blks(NH, TB), TB, 0, stream>>>(agg, (int)NH);
        if (i % 2 == 0) {
            k_scatter64<true><<<blks((long)N_EDGES * 8, TB), TB, 0, stream>>>(
                hcur, ea, we + (long)eix * F_EDGE * HID, be + (long)eix * HID, src, dst, agg);
        } else {
            k_scatter64<false><<<blks((long)N_EDGES * 8, TB), TB, 0, stream>>>(
                hcur, ea, nullptr, nullptr, src, dst, agg);
        }
        k_mm64_wmma<true, true><<<mm_blocks, TB, 0, stream>>>(
            hcur, agg, w1 + (long)i * HID * HID, b1 + (long)i * HID, tmp, N_NODES);
        k_mm64_wmma<false, false><<<mm_blocks, TB, 0, stream>>>(
            tmp, nullptr, w2 + (long)i * HID * HID, b2 + (long)i * HID, hnext, N_NODES);
        k_zero<<<1, 128, 0, stream>>>(stats, 128);
        k_bn_stats<<<blks(N_NODES, 1024), TB, 0, stream>>>(hnext, stats);
        k_bn_apply<<<blks(NH, TB), TB, 0, stream>>>(hnext, stats, gam + (long)(i + 1) * HID,
                                                    bet + (long)(i + 1) * HID);
        float* t = hcur; hcur = hnext; hnext = t;
    }

    // ---- pooling + classifier ----
    k_zero<<<blks((long)N_GRAPHS * 128, TB), TB, 0, stream>>>(pooled, N_GRAPHS * 128);
    k_pool<<<blks(NH, TB), TB, 0, stream>>>(hcur, batch, pooled);
    k_classify<<<blks((long)N_GRAPHS * N_TASKS, TB), TB, 0, stream>>>(pooled, wc, bc, out);
}